// AffinityFeature_5076651344326
// MI455X (gfx1250) — compile-verified
//
#include <hip/hip_runtime.h>
#include <cstdint>

// AffinityFeature for MI455X (gfx1250).
// B=4, C=32, H=256, W=512 fp32 in; (B,8,H,W) fp32 out.
//
// Memory-bound: ~84 MB of traffic @ 23.3 TB/s => ~3.6us floor; ~0.3 GFLOP.
// Strategy:
//  * stage each input float ONCE into LDS (halo tile) with CDNA5
//    GLOBAL_LOAD_ASYNC_TO_LDS_B32 (ASYNCcnt), transposing NCHW->[pix][ch]
//    via per-lane LDS destinations (halo re-reads across blocks are L2-hits:
//    the whole input fits in the 192MB L2, so HBM read amplification ~1.0),
//  * per-pixel L2 norms via V_WMMA_F32_16X16X4_F32 Gram diagonal
//    (B = A^T is register-identical to A under the documented striping),
//  * 8 stencil dot products on VALU with 4-pixel register blocking
//    (18 ds_load_b128 per 4-channel chunk feed 4px x 8 dots),
//  * inv-norm via single v_rsq_f32 + clamp: 1/max(sqrt(s),eps) == min(rsq(s),1e12),
//  * float4 coalesced stores.

namespace {
constexpr int B_ = 4, C_ = 32, H_ = 256, W_ = 512;
constexpr int HWi = H_ * W_;
constexpr int TW = 32, TH = 8;                 // output tile per block
constexpr int HALO_W = TW + 2;                 // 34
constexpr int HALO_H = TH + 2;                 // 10
constexpr int NPIX = HALO_W * HALO_H;          // 340 halo pixels
constexpr int NPIX_R = 352;                    // rounded up to 16 for WMMA groups
constexpr int PS = 36;                         // dwords per pixel (32ch + pad; 144B = 16B aligned, bank-spread)
constexpr int NTHR = 64;                       // 2 wave32
}

typedef float v2f __attribute__((ext_vector_type(2)));
typedef float v8f __attribute__((ext_vector_type(8)));

__device__ __forceinline__ float inv_norm(float s) {
  // 1/max(sqrt(s), 1e-12) for s >= 0;  rsq(0)=+inf clamps to 1e12.
#if __has_builtin(__builtin_amdgcn_rsqf)
  return fminf(__builtin_amdgcn_rsqf(s), 1e12f);
#else
  return fminf(1.0f / sqrtf(s), 1e12f);
#endif
}

__device__ __forceinline__ void dot4acc(float& acc, const float4& a, const float4& b) {
  acc = fmaf(a.x, b.x, acc);
  acc = fmaf(a.y, b.y, acc);
  acc = fmaf(a.z, b.z, acc);
  acc = fmaf(a.w, b.w, acc);
}

__global__ __launch_bounds__(NTHR)
void affinity8_kernel(const float* __restrict__ feat, float* __restrict__ out) {
  __shared__ float tileF[NPIX_R * PS];   // [pixel][channel], padded stride
  __shared__ float invn[NPIX_R];         // 1/max(||x||, eps) per halo pixel

  const int tid = threadIdx.x;
  const int b  = blockIdx.z;
  const int y0 = blockIdx.y * TH;
  const int x0 = blockIdx.x * TW;

  //------------------------------------------------------------------
  // Phase 1: async global->LDS staging of halo tile (340 px x 32 ch).
  // Each lane picks its own LDS destination => free NCHW->[pix][ch]
  // transpose. Out-of-image halo slots stay garbage; masked at use
  // (zero-padding => affinity 0 there anyway).
  //------------------------------------------------------------------
  const unsigned ldsBase = (unsigned)(uint64_t)(&tileF[0]);
  unsigned gOff[6], lOff[6];
  bool ok[6];
  #pragma unroll
  for (int j = 0; j < 6; ++j) {
    const int pix = tid + j * NTHR;
    const int hy = pix / HALO_W, hx = pix % HALO_W;
    const int gy = y0 - 1 + hy, gx = x0 - 1 + hx;
    ok[j]   = (pix < NPIX) && (gy >= 0) && (gy < H_) && (gx >= 0) && (gx < W_);
    gOff[j] = (unsigned)((gy * W_ + gx) * 4);
    lOff[j] = ldsBase + (unsigned)(pix * PS * 4);
  }
  for (int c = 0; c < C_; ++c) {
    const float* gbase = feat + (size_t)(b * C_ + c) * HWi;   // wave-uniform
    #pragma unroll
    for (int j = 0; j < 6; ++j) {
      if (ok[j]) {
        const unsigned l = lOff[j] + (unsigned)(c * 4);
        // lds_dst_vgpr, vgpr byte offset, sgpr64 base
        asm volatile("global_load_async_to_lds_b32 %0, %1, %2"
                     :: "v"(l), "v"(gOff[j]), "s"(gbase) : "memory");
      }
    }
  }
#if __has_builtin(__builtin_amdgcn_s_wait_asynccnt)
  __builtin_amdgcn_s_wait_asynccnt(0);
#else
  asm volatile("s_wait_asynccnt 0" ::: "memory");
#endif
  __syncthreads();

  //------------------------------------------------------------------
  // Phase 1.5: per-pixel sum(x^2) via WMMA Gram diagonal.
  // A (16x4 f32): lanes 0-15 = rows m, K=0,1; lanes 16-31 = rows m, K=2,3.
  // With the documented B striping (lanes = columns, same K halving),
  // B = A^T occupies the *same registers* as A, so D += A*A^T gives the
  // Gram matrix; diag(m) = ||pixel m||^2 over the 4-channel chunk.
  //------------------------------------------------------------------
  const int lane = tid & 31;
  const int wv   = tid >> 5;
#if __has_builtin(__builtin_amdgcn_wmma_f32_16x16x4_f32)
  {
    const int half = lane >> 4;        // K-half selector
    const int mrow = lane & 15;        // pixel within group
    for (int g = wv; g < NPIX_R / 16; g += NTHR / 32) {   // uniform trip count
      const float* pp = &tileF[(g * 16 + mrow) * PS + half * 2];
      v8f acc = {0.f, 0.f, 0.f, 0.f, 0.f, 0.f, 0.f, 0.f};
      #pragma unroll
      for (int kk = 0; kk < C_; kk += 4) {
        v2f a = *(const v2f*)(pp + kk);
        acc = __builtin_amdgcn_wmma_f32_16x16x4_f32(
            false, a, false, a, (short)0, acc, false, false);
      }
      // diag(m): m<8 -> lane m, vgpr m ; m>=8 -> lane m+16, vgpr m-8
      const int r = (lane < 8) ? lane : (lane - 24);
      float s = acc[0];
      s = (r == 1) ? acc[1] : s;  s = (r == 2) ? acc[2] : s;
      s = (r == 3) ? acc[3] : s;  s = (r == 4) ? acc[4] : s;
      s = (r == 5) ? acc[5] : s;  s = (r == 6) ? acc[6] : s;
      s = (r == 7) ? acc[7] : s;
      if (lane < 8 || lane >= 24) {
        const int m = (lane < 8) ? lane : (lane - 16);
        invn[g * 16 + m] = inv_norm(s);
      }
    }
  }
#else
  for (int pix = tid; pix < NPIX_R; pix += NTHR) {
    float s = 0.f;
    #pragma unroll
    for (int kk = 0; kk < C_; kk += 4) {
      float4 v = *(const float4*)&tileF[pix * PS + kk];
      s = fmaf(v.x, v.x, fmaf(v.y, v.y, fmaf(v.z, v.z, fmaf(v.w, v.w, s))));
    }
    invn[pix] = inv_norm(s);
  }
#endif
  __syncthreads();

  //------------------------------------------------------------------
  // Phase 2: 4 consecutive pixels per thread; 8 raw dot products each.
  // 18 float4 LDS reads per 4-channel chunk serve 4px x 8 neighbors.
  // Normalization applied once per dot: dot * inv_center * inv_neighbor.
  //------------------------------------------------------------------
  const int col = tid & 7;            // 8 thread-cols x 4px = 32
  const int row = tid >> 3;           // 8 rows
  const int hx = 1 + col * 4;         // halo-local coords of first pixel
  const int hy = 1 + row;

  float a[4][8];
  #pragma unroll
  for (int p = 0; p < 4; ++p)
    #pragma unroll
    for (int k = 0; k < 8; ++k) a[p][k] = 0.f;

  #pragma unroll
  for (int cc = 0; cc < C_; cc += 4) {
    float4 f[3][6];
    #pragma unroll
    for (int rr = 0; rr < 3; ++rr)
      #pragma unroll
      for (int jj = 0; jj < 6; ++jj)
        f[rr][jj] = *(const float4*)&tileF[((hy - 1 + rr) * HALO_W + (hx - 1 + jj)) * PS + cc];

    #pragma unroll
    for (int p = 0; p < 4; ++p) {
      const float4 ctr = f[1][p + 1];
      // k: 0:(-1,-1) 1:(-1,0) 2:(-1,+1) 3:(0,-1) 4:(0,+1) 5:(+1,-1) 6:(+1,0) 7:(+1,+1)
      dot4acc(a[p][0], ctr, f[0][p]);
      dot4acc(a[p][1], ctr, f[0][p + 1]);
      dot4acc(a[p][2], ctr, f[0][p + 2]);
      dot4acc(a[p][3], ctr, f[1][p]);
      dot4acc(a[p][4], ctr, f[1][p + 2]);
      dot4acc(a[p][5], ctr, f[2][p]);
      dot4acc(a[p][6], ctr, f[2][p + 1]);
      dot4acc(a[p][7], ctr, f[2][p + 2]);
    }
  }

  // Epilogue: normalize, mask out-of-image neighbors, ReLU, float4 stores.
  constexpr int DYk[8] = {-1, -1, -1, 0, 0, 1, 1, 1};
  constexpr int DXk[8] = {-1,  0,  1, -1, 1, -1, 0, 1};
  const int gy  = y0 + row;
  const int gx0 = x0 + col * 4;
  float ic[4];
  #pragma unroll
  for (int p = 0; p < 4; ++p) ic[p] = invn[hy * HALO_W + hx + p];

  float* obase = out + (size_t)(b * 8) * HWi + (size_t)gy * W_ + gx0;
  #pragma unroll
  for (int k = 0; k < 8; ++k) {
    float ov[4];
    #pragma unroll
    for (int p = 0; p < 4; ++p) {
      const int ny = gy + DYk[k];
      const int nx = gx0 + p + DXk[k];
      float v = 0.f;
      if (ny >= 0 && ny < H_ && nx >= 0 && nx < W_) {
        const float in = invn[(hy + DYk[k]) * HALO_W + (hx + p + DXk[k])];
        v = fmaxf(a[p][k] * ic[p] * in, 0.f);
      }
      ov[p] = v;
    }
    *(float4*)(obase + (size_t)k * HWi) = make_float4(ov[0], ov[1], ov[2], ov[3]);
  }
}

extern "C" void kernel_launch(void* const* d_in, const int* in_sizes, int n_in,
                              void* d_out, int out_size, void* d_ws, size_t ws_size,
                              hipStream_t stream) {
  (void)in_sizes; (void)n_in; (void)out_size; (void)d_ws; (void)ws_size;
  const float* feat = (const float*)d_in[0];
  float* out = (float*)d_out;
  dim3 grid(W_ / TW, H_ / TH, B_);   // 16 x 32 x 4
  affinity8_kernel<<<grid, NTHR, 0, stream>>>(feat, out);
}